// ImplicitGraph_41308995452956
// MI455X (gfx1250) — compile-verified
//
#include <hip/hip_runtime.h>
#include <hip/hip_bf16.h>

typedef __attribute__((ext_vector_type(2))) float v2f;
typedef __attribute__((ext_vector_type(8))) float v8f;

#define MDIM 128
#define NT   16     // node (N) tile per GEMM slab
#define KAPPA_V 0.99f

// ---------------------------------------------------------------------------
// Row-wise projection of W onto L1 ball of radius KAPPA (=> ||W||_inf <= v).
// ---------------------------------------------------------------------------
__global__ __launch_bounds__(128) void proj_kernel(const float* __restrict__ W,
                                                   float* __restrict__ Wp) {
  __shared__ float aw[128];
  __shared__ float srt[128];
  __shared__ float css[128];
  __shared__ float s_alpha;
  __shared__ float s_rowl1;
  const int row = blockIdx.x;
  const int t = threadIdx.x;
  const float w = W[row * MDIM + t];
  const float a = fabsf(w);
  aw[t] = a;
  __syncthreads();
  int rank = 0;
  for (int j = 0; j < MDIM; ++j) {
    float aj = aw[j];
    rank += (aj > a) || (aj == a && j < t);
  }
  srt[rank] = a;
  __syncthreads();
  if (t == 0) {
    float s = 0.0f;
    for (int j = 0; j < MDIM; ++j) { s += srt[j]; css[j] = s - KAPPA_V; }
    s_rowl1 = s;
    int rho = 0;
    for (int j = 0; j < MDIM; ++j)
      rho += (srt[j] * (float)(j + 1) > css[j]) ? 1 : 0;
    s_alpha = css[rho - 1] / (float)rho;
  }
  __syncthreads();
  const float alpha = s_alpha;
  float proj = fmaxf(a - alpha, 0.0f);
  proj = (w > 0.0f) ? proj : ((w < 0.0f) ? -proj : 0.0f);
  Wp[row * MDIM + t] = (s_rowl1 > KAPPA_V) ? proj : w;
}

// ---------------------------------------------------------------------------
// Transpose U (P=128 x N row-major) -> Ut (node-major: Ut[n*128 + p]).
// ---------------------------------------------------------------------------
__global__ __launch_bounds__(256) void transpose_kernel(const float* __restrict__ U,
                                                        float* __restrict__ Ut,
                                                        int N) {
  __shared__ float tile[32][33];
  const int n0 = blockIdx.x * 32;
  const int p0 = blockIdx.y * 32;
  const int tx = threadIdx.x;
  const int ty = threadIdx.y;
  for (int i = ty; i < 32; i += 8) {
    int n = n0 + tx;
    tile[i][tx] = (n < N) ? U[(size_t)(p0 + i) * N + n] : 0.0f;
  }
  __syncthreads();
  for (int i = ty; i < 32; i += 8) {
    int n = n0 + i;
    if (n < N) Ut[(size_t)n * MDIM + p0 + tx] = tile[tx][i];
  }
}

// ---------------------------------------------------------------------------
// Persistent grid-stride GEMM: Y = A(128x128) @ X, X/Y node-major (N x 128).
// A staged in LDS ONCE per block; 16-node X slabs double-buffered so the
// global->LDS load of slab s+1 overlaps the 32-WMMA chain on slab s.
// 8 waves/block; wave w owns rows [16w,16w+16). V_WMMA_F32_16X16X4_F32.
// ---------------------------------------------------------------------------
__global__ __launch_bounds__(256) void gemm_kernel(const float* __restrict__ A,
                                                   const float* __restrict__ X,
                                                   float* __restrict__ Y,
                                                   int N, int nSlabs) {
  __shared__ float As[MDIM][MDIM + 1];        // 66 KB
  __shared__ float Xs[2][NT][MDIM + 1];       // 2 x 8.25 KB
  const int tid = threadIdx.x;

  for (int i = tid; i < MDIM * MDIM; i += 256)
    As[i >> 7][i & 127] = A[i];

  const int wave = tid >> 5;
  const int lane = tid & 31;
  const int half = lane >> 4;                 // lanes 16-31 take K+2 (ISA layout)
  const int l15  = lane & 15;
  const int mrow = wave * 16 + l15;

  int slab = blockIdx.x;
  if (slab < nSlabs) {                        // preload first slab
    for (int i = tid; i < NT * MDIM; i += 256) {
      int nn = i >> 7, k = i & 127;
      int n = slab * NT + nn;
      Xs[0][nn][k] = (n < N) ? X[(size_t)n * MDIM + k] : 0.0f;
    }
  }
  __syncthreads();

  int buf = 0;
  for (; slab < nSlabs; slab += gridDim.x) {
    const int nslab = slab + gridDim.x;       // prefetch next slab -> other buffer
    if (nslab < nSlabs) {
      for (int i = tid; i < NT * MDIM; i += 256) {
        int nn = i >> 7, k = i & 127;
        int n = nslab * NT + nn;
        Xs[1 - buf][nn][k] = (n < N) ? X[(size_t)n * MDIM + k] : 0.0f;
      }
    }

    v8f c = {};
    #pragma unroll
    for (int kb = 0; kb < MDIM / 4; ++kb) {
      const int k0 = kb * 4 + 2 * half;
      v2f a, b;
      a.x = As[mrow][k0];
      a.y = As[mrow][k0 + 1];
      b.x = Xs[buf][l15][k0];
      b.y = Xs[buf][l15][k0 + 1];
      c = __builtin_amdgcn_wmma_f32_16x16x4_f32(false, a, false, b,
                                                (short)0, c, false, false);
    }

    const int n = slab * NT + l15;
    if (n < N) {
      float* outp = Y + (size_t)n * MDIM + wave * 16 + half * 8;
      #pragma unroll
      for (int r = 0; r < 8; ++r) outp[r] = c[r];
    }
    __syncthreads();                          // next buffer now fully loaded
    buf ^= 1;
  }
}

// ---------------------------------------------------------------------------
// CSC build: degree histogram, single-block exclusive scan, cursor fill.
// ---------------------------------------------------------------------------
__global__ __launch_bounds__(256) void count_kernel(const int* __restrict__ ec,
                                                    int* __restrict__ deg, int E) {
  int e = blockIdx.x * 256 + threadIdx.x;
  if (e < E)
    __hip_atomic_fetch_add(&deg[ec[e]], 1, __ATOMIC_RELAXED, __HIP_MEMORY_SCOPE_AGENT);
}

__global__ __launch_bounds__(1024) void scan_kernel(const int* __restrict__ deg,
                                                    int* __restrict__ rowptr, int N) {
  __shared__ int buf[1024];
  __shared__ int carry;
  if (threadIdx.x == 0) carry = 0;
  __syncthreads();
  for (int base = 0; base < N; base += 1024) {
    const int c0 = carry;
    const int i = base + (int)threadIdx.x;
    const int x = (i < N) ? deg[i] : 0;
    buf[threadIdx.x] = x;
    __syncthreads();
    for (int off = 1; off < 1024; off <<= 1) {
      int v = buf[threadIdx.x];
      int add = (threadIdx.x >= (unsigned)off) ? buf[threadIdx.x - off] : 0;
      __syncthreads();
      buf[threadIdx.x] = v + add;
      __syncthreads();
    }
    if (i < N) rowptr[i] = c0 + buf[threadIdx.x] - x;   // exclusive
    __syncthreads();
    if (threadIdx.x == 0) carry = c0 + buf[1023];
    __syncthreads();
  }
  if (threadIdx.x == 0) rowptr[N] = carry;
}

__global__ __launch_bounds__(256) void fill_kernel(const int* __restrict__ er,
                                                   const int* __restrict__ ec,
                                                   const float* __restrict__ ev,
                                                   int* __restrict__ cursor,
                                                   int* __restrict__ csr_row,
                                                   float* __restrict__ csr_val, int E) {
  int e = blockIdx.x * 256 + threadIdx.x;
  if (e >= E) return;
  int pos = __hip_atomic_fetch_add(&cursor[ec[e]], 1, __ATOMIC_RELAXED,
                                   __HIP_MEMORY_SCOPE_AGENT);
  csr_row[pos] = er[e];
  csr_val[pos] = ev[e];
}

// ---------------------------------------------------------------------------
// Fused A^T-gather (+B, +ReLU): one wave per node.
//   Xout[:,n] = act( Badd[:,n] + sum_e val_e * Y[:,row_e] )
// ---------------------------------------------------------------------------
__global__ __launch_bounds__(256) void gather_kernel(const int* __restrict__ rowptr,
                                                     const int* __restrict__ csr_row,
                                                     const float* __restrict__ csr_val,
                                                     const float* __restrict__ Y,
                                                     const float* __restrict__ Badd,
                                                     float* __restrict__ Xout,
                                                     int N, int doRelu) {
  const int n = (int)((blockIdx.x * 256 + threadIdx.x) >> 5);
  if (n >= N) return;
  const int lane = threadIdx.x & 31;
  const int start = rowptr[n];
  const int end   = rowptr[n + 1];

  float4 acc = make_float4(0.0f, 0.0f, 0.0f, 0.0f);
  if (Badd) acc = *(const float4*)(Badd + (size_t)n * MDIM + lane * 4);

  for (int i = start; i < end; i += 32) {
    const int cnt = min(32, end - i);
    const int   rr = (lane < cnt) ? csr_row[i + lane] : 0;
    const float vv = (lane < cnt) ? csr_val[i + lane] : 0.0f;
    for (int j = 0; j < cnt; ++j) {
      const int   r = __shfl(rr, j, 32);
      const float v = __shfl(vv, j, 32);
      if (j + 1 < cnt) {  // prefetch next source column
        const int rn = __shfl(rr, j + 1, 32);
        __builtin_prefetch(Y + (size_t)rn * MDIM + lane * 4, 0, 1);
      }
      const float4 y = *(const float4*)(Y + (size_t)r * MDIM + lane * 4);
      acc.x += v * y.x; acc.y += v * y.y; acc.z += v * y.z; acc.w += v * y.w;
    }
  }
  if (doRelu) {
    acc.x = fmaxf(acc.x, 0.0f); acc.y = fmaxf(acc.y, 0.0f);
    acc.z = fmaxf(acc.z, 0.0f); acc.w = fmaxf(acc.w, 0.0f);
  }
  *(float4*)(Xout + (size_t)n * MDIM + lane * 4) = acc;
}

// ---------------------------------------------------------------------------
// Final: out[m*N + n] = P[n*128 + m]  (node-major -> row-major (M,N)).
// ---------------------------------------------------------------------------
__global__ __launch_bounds__(256) void out_kernel(const float* __restrict__ P,
                                                  float* __restrict__ out, int N) {
  __shared__ float tile[32][33];
  const int n0 = blockIdx.x * 32;
  const int m0 = blockIdx.y * 32;
  const int tx = threadIdx.x;
  const int ty = threadIdx.y;
  for (int i = ty; i < 32; i += 8) {
    int n = n0 + i;
    tile[i][tx] = (n < N) ? P[(size_t)n * MDIM + m0 + tx] : 0.0f;
  }
  __syncthreads();
  for (int i = ty; i < 32; i += 8) {
    int n = n0 + tx;
    if (n < N) out[(size_t)(m0 + i) * N + n] = tile[tx][i];
  }
}

// ---------------------------------------------------------------------------
extern "C" void kernel_launch(void* const* d_in, const int* in_sizes, int n_in,
                              void* d_out, int out_size, void* d_ws, size_t ws_size,
                              hipStream_t stream) {
  const float* W   = (const float*)d_in[0];
  const float* Om  = (const float*)d_in[1];
  const float* U   = (const float*)d_in[2];
  const float* ev  = (const float*)d_in[3];
  const int*   er  = (const int*)d_in[4];
  const int*   ec  = (const int*)d_in[5];
  const int N = in_sizes[2] / MDIM;
  const int E = in_sizes[3];
  float* out = (float*)d_out;

  const size_t mn = (size_t)MDIM * N;

  // workspace layout
  float* Wp   = (float*)d_ws;                 // 16384 floats
  float* Y    = Wp + 16384;                   // mn
  float* Bb   = Y  + mn;                      // mn
  float* X0   = Bb + mn;                      // mn
  float* X1   = X0 + mn;                      // mn
  float* csr_val = X1 + mn;                   // E floats
  int*   csr_row = (int*)(csr_val + E);       // E ints
  int*   rowptr  = csr_row + E;               // N+1 ints
  int*   cursor  = rowptr + (N + 1);          // N ints (also degree histogram)

  const int nSlabs     = (N + NT - 1) / NT;
  const int gemmBlocks = (nSlabs < 1024) ? nSlabs : 1024;   // persistent blocks
  const int tr32       = (N + 31) / 32;
  const int edgeBlocks = (E + 255) / 256;
  const int nodeBlocks = (N + 7) / 8;         // one wave per node

  // --- one-time setup -----------------------------------------------------
  proj_kernel<<<128, 128, 0, stream>>>(W, Wp);
  transpose_kernel<<<dim3(tr32, MDIM / 32), dim3(32, 8), 0, stream>>>(U, X1, N);
  // CSC of A (indexed by destination col)
  hipMemsetAsync(cursor, 0, (size_t)N * sizeof(int), stream);
  count_kernel<<<edgeBlocks, 256, 0, stream>>>(ec, cursor, E);
  scan_kernel<<<1, 1024, 0, stream>>>(cursor, rowptr, N);
  hipMemcpyAsync(cursor, rowptr, (size_t)N * sizeof(int),
                 hipMemcpyDeviceToDevice, stream);
  fill_kernel<<<edgeBlocks, 256, 0, stream>>>(er, ec, ev, cursor, csr_row, csr_val, E);
  // B = A^T-gather(Omega_1 @ Ut)    (no bias, no relu)
  gemm_kernel<<<gemmBlocks, 256, 0, stream>>>(Om, X1, Y, N, nSlabs);
  gather_kernel<<<nodeBlocks, 256, 0, stream>>>(rowptr, csr_row, csr_val, Y,
                                                (const float*)nullptr, Bb, N, 0);

  // --- fixed-point loop: X_{k+1} = relu(A^T (Wp X_k) + B), X_0 = B --------
  const float* cur = Bb;
  float* nxt = X0;
  const int ITERS = 51;   // 50 iterations + final recompute (Z = relu(step+B))
  for (int it = 0; it < ITERS; ++it) {
    gemm_kernel<<<gemmBlocks, 256, 0, stream>>>(Wp, cur, Y, N, nSlabs);
    gather_kernel<<<nodeBlocks, 256, 0, stream>>>(rowptr, csr_row, csr_val, Y,
                                                  Bb, nxt, N, 1);
    cur = nxt;
    nxt = (nxt == X0) ? X1 : X0;
  }
  out_kernel<<<dim3(tr32, MDIM / 32), dim3(32, 8), 0, stream>>>(cur, out, N);
}